// GNN_79405355368572
// MI455X (gfx1250) — compile-verified
//
#include <hip/hip_runtime.h>
#include <hip/hip_bf16.h>
#include <math.h>

#define NB 256
#define GG 400
#define DEG 32
#define NN (NB * GG)          // 102400 nodes
#define EE (NN * DEG)         // 3276800 edges
#define ETOT (EE + NN)        // + self loops
#define INDIM 400
#define D1 32
#define D2 8
#define NCLASS 2
#define KSEL 200
#define SLOPE 0.2f
#define BN_EPS 1e-5f

#define KC 8                  // K-chunk per async stage (2 wmma k-steps)
#define NCHUNK (INDIM / KC)   // 50
#define XROW 12               // padded LDS row stride (floats): 48B -> 16B aligned, bank-conflict free

typedef __attribute__((ext_vector_type(2))) float v2f;
typedef __attribute__((ext_vector_type(8))) float v8f;
// match the builtin's parameter types exactly: GCC-vector int4, AS1 (global) / AS3 (LDS)
typedef int v4i __attribute__((vector_size(16)));
typedef __attribute__((address_space(1))) v4i gv4i;
typedef __attribute__((address_space(3))) v4i lv4i;

// ---- order-preserving float<->uint encoding (for atomicMax on floats) ----
__device__ __forceinline__ unsigned f2ord(float f) {
    unsigned u = __float_as_uint(f);
    return (u >> 31) ? ~u : (u | 0x80000000u);
}
__device__ __forceinline__ float ord2f(unsigned o) {
    return (o & 0x80000000u) ? __uint_as_float(o ^ 0x80000000u) : __uint_as_float(~o);
}

// =====================================================================
// 1) h = x @ W  via V_WMMA_F32_16X16X4_F32, with x staged through LDS by
//    GLOBAL_LOAD_ASYNC_TO_LDS_B128 (ASYNCcnt), double-buffered.
//    Block = 256 threads (8 waves) -> 128 rows; grid = 800 blocks.
// =====================================================================
__global__ __launch_bounds__(256) void k_gemm_h(const float* __restrict__ x,
                                                const float* __restrict__ W,
                                                float* __restrict__ h) {
    __shared__ float Wl[INDIM * D1];        // 50 KB
    __shared__ float Xl[2][128 * XROW];     // 2 x 6 KB

    for (int i = threadIdx.x; i < INDIM * D1; i += 256) Wl[i] = W[i];

    const int lane = threadIdx.x & 31;
    const int wave = threadIdx.x >> 5;
    const int m    = lane & 15;             // row (A) / col (B,D) within tile
    const int half = lane >> 4;             // K-pair selector
    const size_t row0 = (size_t)blockIdx.x * 128 + (size_t)wave * 16;

    // async-copy geometry: 256 threads x 16B = one 128x8 f32 chunk
    const int trow  = threadIdx.x >> 1;         // 0..127
    const int tcol4 = (threadIdx.x & 1) * 4;    // 0 or 4
    const float* gbase = x + ((size_t)blockIdx.x * 128 + trow) * INDIM + tcol4;
    float* lbase = &Xl[0][0] + trow * XROW + tcol4;

    // issue chunk 0
    __builtin_amdgcn_global_load_async_to_lds_b128((gv4i*)gbase, (lv4i*)lbase, 0, 0);

    v8f acc0 = {};   // cols 0..15
    v8f acc1 = {};   // cols 16..31

    for (int c = 0; c < NCHUNK; ++c) {
        const int buf = c & 1;
        if (c + 1 < NCHUNK) {
            __builtin_amdgcn_global_load_async_to_lds_b128(
                (gv4i*)(gbase + (size_t)(c + 1) * KC),
                (lv4i*)(lbase + ((c + 1) & 1) * (128 * XROW)), 0, 0);
            asm volatile("s_wait_asynccnt 0x1" ::: "memory");   // chunk c landed (in-order)
        } else {
            asm volatile("s_wait_asynccnt 0x0" ::: "memory");   // last chunk landed
        }
        __syncthreads();   // all waves' async writes (and Wl on c==0) visible

        const float* xl = &Xl[buf][0] + (wave * 16 + m) * XROW;
        #pragma unroll
        for (int kk = 0; kk < KC; kk += 4) {
            const int ka = kk + 2 * half;          // K within chunk for this lane half
            v2f a; a.x = xl[ka]; a.y = xl[ka + 1];
            const int kg = c * KC + ka;            // global K for B fragments
            v2f b0; b0.x = Wl[kg * D1 + m];       b0.y = Wl[(kg + 1) * D1 + m];
            v2f b1; b1.x = Wl[kg * D1 + 16 + m];  b1.y = Wl[(kg + 1) * D1 + 16 + m];
            acc0 = __builtin_amdgcn_wmma_f32_16x16x4_f32(false, a, false, b0, (short)0, acc0, false, false);
            acc1 = __builtin_amdgcn_wmma_f32_16x16x4_f32(false, a, false, b1, (short)0, acc1, false, false);
        }
        __syncthreads();   // protect buf before chunk c+2 overwrites it
    }

#pragma unroll
    for (int r = 0; r < 8; ++r) {
        // D layout: VGPR r -> M = r (lanes 0-15) / r+8 (lanes 16-31), N = lane%16
        const size_t row = row0 + (size_t)(r + 8 * half);
        h[row * D1 + m]       = acc0[r];
        h[row * D1 + 16 + m]  = acc1[r];
    }
}

// =====================================================================
// 2) per-node attention scalars + workspace init (max/denom/out)
// =====================================================================
__global__ void k_node_init(const float* __restrict__ h,
                            const float* __restrict__ att_src,
                            const float* __restrict__ att_dst,
                            float* __restrict__ a_src, float* __restrict__ a_dst,
                            unsigned* __restrict__ mord, float* __restrict__ denom,
                            float* __restrict__ out) {
    const int i = blockIdx.x * blockDim.x + threadIdx.x;
    if (i >= NN) return;
    const float* hr = h + (size_t)i * D1;
    float as = 0.f, ad = 0.f;
#pragma unroll
    for (int f = 0; f < D1; ++f) { as += hr[f] * att_src[f]; ad += hr[f] * att_dst[f]; }
    a_src[i] = as; a_dst[i] = ad;
    mord[i]  = 0u;        // encodes "below any real value"
    denom[i] = 0.f;
    float* orow = out + (size_t)i * D1;
#pragma unroll
    for (int f = 0; f < D1; ++f) orow[f] = 0.f;
}

// =====================================================================
// 3) segment-max over incoming edges (includes self loops)
// =====================================================================
__global__ void k_edge_max(const int* __restrict__ ei,
                           const float* __restrict__ a_src,
                           const float* __restrict__ a_dst,
                           unsigned* __restrict__ mord) {
    const long long t = (long long)blockIdx.x * blockDim.x + threadIdx.x;
    if (t >= ETOT) return;
    int s, d;
    if (t < EE) { s = ei[t]; d = ei[EE + t]; } else { s = d = (int)(t - EE); }
    float e = a_src[s] + a_dst[d];
    e = (e > 0.f) ? e : SLOPE * e;
    atomicMax(&mord[d], f2ord(e));
}

// =====================================================================
// 4) exp(e - max) + denominator accumulation
// =====================================================================
__global__ void k_edge_exp(const int* __restrict__ ei,
                           const float* __restrict__ a_src,
                           const float* __restrict__ a_dst,
                           const unsigned* __restrict__ mord,
                           float* __restrict__ ew, float* __restrict__ denom) {
    const long long t = (long long)blockIdx.x * blockDim.x + threadIdx.x;
    if (t >= ETOT) return;
    int s, d;
    if (t < EE) { s = ei[t]; d = ei[EE + t]; } else { s = d = (int)(t - EE); }
    float e = a_src[s] + a_dst[d];
    e = (e > 0.f) ? e : SLOPE * e;
    const float w = __expf(e - ord2f(mord[d]));
    ew[t] = w;
    atomicAdd(&denom[d], w);
}

// =====================================================================
// 5) weighted aggregation: out[d] += alpha * h[s]
// =====================================================================
__global__ void k_edge_aggr(const int* __restrict__ ei,
                            const float* __restrict__ ew,
                            const float* __restrict__ denom,
                            const float* __restrict__ h,
                            float* __restrict__ out) {
    const long long t = (long long)blockIdx.x * blockDim.x + threadIdx.x;
    if (t >= ETOT) return;
    int s, d;
    if (t < EE) { s = ei[t]; d = ei[EE + t]; } else { s = d = (int)(t - EE); }
    const float alpha = ew[t] / denom[d];
    const float* hs = h + (size_t)s * D1;
    float* od = out + (size_t)d * D1;
#pragma unroll
    for (int f = 0; f < D1; ++f) atomicAdd(&od[f], alpha * hs[f]);
}

// =====================================================================
// 6) TopK(ratio=0.5) per graph + fused global max/mean pooling.
//    One block (256 thr) per graph; bitonic sort of 512 packed keys.
// =====================================================================
__global__ __launch_bounds__(256) void k_topk(const float* __restrict__ out,
                                              const float* __restrict__ gbias,
                                              const float* __restrict__ poolw,
                                              float* __restrict__ z,
                                              float* __restrict__ d_vals,
                                              float* __restrict__ d_perm) {
    __shared__ unsigned long long keys[512];
    const int b = blockIdx.x;
    const int t = threadIdx.x;

    float nrm = 0.f;
#pragma unroll
    for (int f = 0; f < D1; ++f) nrm += poolw[f] * poolw[f];
    nrm = sqrtf(nrm);

    for (int s = 0; s < 2; ++s) {
        const int i = t + s * 256;
        unsigned long long key = ~0ull;          // padding sorts last
        if (i < GG) {
            const float* orow = out + ((size_t)b * GG + i) * D1;
            float sc = 0.f;
#pragma unroll
            for (int f = 0; f < D1; ++f) sc += (orow[f] + gbias[f]) * poolw[f];
            sc /= nrm;
            const float sig = 1.f / (1.f + __expf(-sc));
            // sort ascending == value descending, index ascending on ties
            key = ((unsigned long long)(~f2ord(sig)) << 32) | (unsigned)i;
        }
        keys[i] = key;
    }
    __syncthreads();

    for (int k = 2; k <= 512; k <<= 1) {
        for (int j = k >> 1; j > 0; j >>= 1) {
            for (int s = 0; s < 2; ++s) {
                const int i = t + s * 256;
                const int ixj = i ^ j;
                if (ixj > i) {
                    const bool up = ((i & k) == 0);
                    const unsigned long long a = keys[i], c = keys[ixj];
                    if ((a > c) == up) { keys[i] = c; keys[ixj] = a; }
                }
            }
            __syncthreads();
        }
    }

    if (t < KSEL) {
        const unsigned long long key = keys[t];
        const unsigned idx = (unsigned)(key & 0xFFFFFFFFull);
        const float sig = ord2f(~(unsigned)(key >> 32));
        d_vals[b * KSEL + t] = 1.f / (1.f + __expf(-sig));   // sigmoid(sigmoid(score))
        d_perm[b * KSEL + t] = (float)(b * GG + (int)idx);
    }

    if (t < D1) {
        const int f = t;
        float mx = -INFINITY, sm = 0.f;
        for (int j = 0; j < KSEL; ++j) {
            const unsigned long long key = keys[j];
            const unsigned idx = (unsigned)(key & 0xFFFFFFFFull);
            const float sig = ord2f(~(unsigned)(key >> 32));
            const float feat = (out[((size_t)b * GG + idx) * D1 + f] + gbias[f]) * sig;
            mx = fmaxf(mx, feat);
            sm += feat;
        }
        z[b * (2 * D1) + f]      = mx;
        z[b * (2 * D1) + D1 + f] = sm / (float)KSEL;
    }
}

// =====================================================================
// 7) MLP tail + log_softmax; one thread per graph. Also copies pool_w.
// =====================================================================
__global__ __launch_bounds__(256) void k_mlp(const float* __restrict__ z,
                                             const float* __restrict__ fc1w, const float* __restrict__ fc1b,
                                             const float* __restrict__ bn1g, const float* __restrict__ bn1b,
                                             const float* __restrict__ fc2w, const float* __restrict__ fc2b,
                                             const float* __restrict__ bn2g, const float* __restrict__ bn2b,
                                             const float* __restrict__ fc3w, const float* __restrict__ fc3b,
                                             const float* __restrict__ poolw,
                                             float* __restrict__ logits, float* __restrict__ poolw_out) {
    const int b = threadIdx.x;
    if (b < D1) poolw_out[b] = poolw[b];

    const float rsq = 1.f / sqrtf(1.f + BN_EPS);
    const float* zr = z + b * (2 * D1);

    float z1[D1];
    for (int o = 0; o < D1; ++o) {
        float acc = fc1b[o];
        for (int i = 0; i < 2 * D1; ++i) acc += zr[i] * fc1w[i * D1 + o];
        acc = fmaxf(acc, 0.f);
        z1[o] = acc * (bn1g[o] * rsq) + bn1b[o];
    }
    float z2[D2];
    for (int o = 0; o < D2; ++o) {
        float acc = fc2b[o];
        for (int i = 0; i < D1; ++i) acc += z1[i] * fc2w[i * D2 + o];
        acc = fmaxf(acc, 0.f);
        z2[o] = acc * (bn2g[o] * rsq) + bn2b[o];
    }
    float l0 = fc3b[0], l1 = fc3b[1];
    for (int i = 0; i < D2; ++i) { l0 += z2[i] * fc3w[i * NCLASS]; l1 += z2[i] * fc3w[i * NCLASS + 1]; }
    const float mx = fmaxf(l0, l1);
    const float lse = mx + logf(__expf(l0 - mx) + __expf(l1 - mx));
    logits[b * NCLASS + 0] = l0 - lse;
    logits[b * NCLASS + 1] = l1 - lse;
}

// =====================================================================
extern "C" void kernel_launch(void* const* d_in, const int* in_sizes, int n_in,
                              void* d_out, int out_size, void* d_ws, size_t ws_size,
                              hipStream_t stream) {
    const float* x       = (const float*)d_in[0];
    const int*   ei      = (const int*)  d_in[1];
    // d_in[2] = batch (unused: derivable), d_in[3] = edge_attr (unused by reference)
    const float* W       = (const float*)d_in[4];
    const float* att_src = (const float*)d_in[5];
    const float* att_dst = (const float*)d_in[6];
    const float* gbias   = (const float*)d_in[7];
    const float* poolw   = (const float*)d_in[8];
    const float* fc1w    = (const float*)d_in[9];
    const float* fc1b    = (const float*)d_in[10];
    const float* bn1g    = (const float*)d_in[11];
    const float* bn1b    = (const float*)d_in[12];
    const float* fc2w    = (const float*)d_in[13];
    const float* fc2b    = (const float*)d_in[14];
    const float* bn2g    = (const float*)d_in[15];
    const float* bn2b    = (const float*)d_in[16];
    const float* fc3w    = (const float*)d_in[17];
    const float* fc3b    = (const float*)d_in[18];

    // workspace carve-up
    float*    h     = (float*)d_ws;                       // NN*D1
    float*    a_src = h + (size_t)NN * D1;                // NN
    float*    a_dst = a_src + NN;                         // NN
    unsigned* mord  = (unsigned*)(a_dst + NN);            // NN
    float*    denom = (float*)(mord + NN);                // NN
    float*    out   = denom + NN;                         // NN*D1
    float*    ew    = out + (size_t)NN * D1;              // ETOT
    float*    z     = ew + (size_t)ETOT;                  // NB*64

    // output packing: (logits, pool_w, sigmoid(vals), perm)
    float* o_logits = (float*)d_out;                      // NB*NCLASS
    float* o_poolw  = o_logits + NB * NCLASS;             // D1
    float* o_vals   = o_poolw + D1;                       // NB*KSEL
    float* o_perm   = o_vals + NB * KSEL;                 // NB*KSEL

    const int egrid = (ETOT + 255) / 256;

    k_gemm_h  <<<NN / 128, 256, 0, stream>>>(x, W, h);
    k_node_init<<<(NN + 255) / 256, 256, 0, stream>>>(h, att_src, att_dst,
                                                      a_src, a_dst, mord, denom, out);
    k_edge_max <<<egrid, 256, 0, stream>>>(ei, a_src, a_dst, mord);
    k_edge_exp <<<egrid, 256, 0, stream>>>(ei, a_src, a_dst, mord, ew, denom);
    k_edge_aggr<<<egrid, 256, 0, stream>>>(ei, ew, denom, h, out);
    k_topk     <<<NB, 256, 0, stream>>>(out, gbias, poolw, z, o_vals, o_perm);
    k_mlp      <<<1, 256, 0, stream>>>(z, fc1w, fc1b, bn1g, bn1b,
                                       fc2w, fc2b, bn2g, bn2b, fc3w, fc3b,
                                       poolw, o_logits, o_poolw);
}